// Model_36301063586078
// MI455X (gfx1250) — compile-verified
//
#include <hip/hip_runtime.h>
#include <hip/hip_bf16.h>

typedef __attribute__((ext_vector_type(2))) float v2f;
typedef __attribute__((ext_vector_type(8))) float v8f;

#define N_NODES 100000
#define N_EDGES 1600000
#define N_LABEL 500000

// ---------------------------------------------------------------------------
// zero a float buffer (grid-stride)
// ---------------------------------------------------------------------------
__global__ __launch_bounds__(256) void zero_f32(float* __restrict__ p, long n) {
    long i = (long)blockIdx.x * blockDim.x + threadIdx.x;
    long stride = (long)gridDim.x * blockDim.x;
    for (; i < n; i += stride) p[i] = 0.0f;
}

// ---------------------------------------------------------------------------
// scatter: one wave (32 lanes) per edge; each lane adds a float2 of the
// 64-float source feature row into agg[dst]. Lane 0 optionally bumps cnt.
// ---------------------------------------------------------------------------
__global__ __launch_bounds__(256) void edge_scatter(
    const float* __restrict__ feat, const int* __restrict__ src,
    const int* __restrict__ dst, float* __restrict__ agg,
    float* __restrict__ cnt, int nEdges, int doCnt)
{
    int wid  = (int)(((long)blockIdx.x * blockDim.x + threadIdx.x) >> 5);
    int lane = threadIdx.x & 31;
    if (wid >= nEdges) return;
    int s = src[wid];
    int d = dst[wid];
    const v2f* fp = (const v2f*)(feat + (size_t)s * 64);
    v2f v = fp[lane];
    float* ap = agg + (size_t)d * 64 + lane * 2;
    atomicAdd(ap + 0, v.x);
    atomicAdd(ap + 1, v.y);
    if (doCnt && lane == 0) atomicAdd(&cnt[d], 1.0f);
}

// ---------------------------------------------------------------------------
// SAGE combine: out = act( (agg/max(cnt,1)) @ Wl + bl + xin @ Wr )
// 128 threads = 4 waves; block handles a 16-node tile; wave w computes the
// 16x16 output tile for columns [16w, 16w+16). K=64 per GEMM, f32 WMMA.
// A layout (16x4 f32): lane&15 = M row, VGPR0/1 = K { half*2, half*2+1 }.
// ---------------------------------------------------------------------------
__global__ __launch_bounds__(128) void sage_combine(
    const float* __restrict__ agg, const float* __restrict__ cnt,
    const float* __restrict__ xin, const float* __restrict__ Wl,
    const float* __restrict__ bl,  const float* __restrict__ Wr,
    float* __restrict__ out, int doRelu)
{
    const int lane = threadIdx.x & 31;
    const int wave = threadIdx.x >> 5;     // 0..3 -> output column tile
    const int tile = blockIdx.x;           // 16-node tile
    const int r    = lane & 15;
    const int half = lane >> 4;            // 0: K{0,1}, 1: K{2,3}
    const int row  = tile * 16 + r;        // A-matrix row for this lane
    const int col  = wave * 16 + r;        // B-matrix column for this lane

    const float inv = 1.0f / fmaxf(cnt[row], 1.0f);

    const float* arow = agg + (size_t)row * 64 + half * 2;
    const float* xrow = xin + (size_t)row * 64 + half * 2;

    v8f acc = {};
#pragma unroll
    for (int k0 = 0; k0 < 64; k0 += 4) {
        const int kk = k0 + half * 2;
        v2f a, b;
        a.x = arow[k0]     * inv;
        a.y = arow[k0 + 1] * inv;
        b.x = Wl[kk * 64 + col];
        b.y = Wl[(kk + 1) * 64 + col];
        acc = __builtin_amdgcn_wmma_f32_16x16x4_f32(false, a, false, b,
                                                    (short)0, acc, false, false);
    }
#pragma unroll
    for (int k0 = 0; k0 < 64; k0 += 4) {
        const int kk = k0 + half * 2;
        v2f a, b;
        a.x = xrow[k0];
        a.y = xrow[k0 + 1];
        b.x = Wr[kk * 64 + col];
        b.y = Wr[(kk + 1) * 64 + col];
        acc = __builtin_amdgcn_wmma_f32_16x16x4_f32(false, a, false, b,
                                                    (short)0, acc, false, false);
    }

    const float bias = bl[col];
#pragma unroll
    for (int i = 0; i < 8; ++i) {
        float v = acc[i] + bias;
        if (doRelu) v = fmaxf(v, 0.0f);
        const int m = i + half * 8;        // C layout: row m, column = col
        out[(size_t)(tile * 16 + m) * 64 + col] = v;
    }
}

// ---------------------------------------------------------------------------
// Decoder: h = relu(concat(z[row], z[col]) @ Wd1 + bd1); out = h @ Wd2 + bd2
// 128 threads = 4 waves; 16 label pairs per block. K=128 split into two
// K=64 WMMA chains (row-half then col-half of the concat).
// ---------------------------------------------------------------------------
__global__ __launch_bounds__(128) void link_decoder(
    const float* __restrict__ z, const int* __restrict__ rowi,
    const int* __restrict__ coli, const float* __restrict__ Wd1,
    const float* __restrict__ bd1, const float* __restrict__ Wd2,
    const float* __restrict__ bd2, float* __restrict__ out)
{
    __shared__ float hbuf[16][68];         // 16 pairs x 64 hidden (+pad)

    const int lane = threadIdx.x & 31;
    const int wave = threadIdx.x >> 5;
    const int tile = blockIdx.x;
    const int r    = lane & 15;
    const int half = lane >> 4;
    const int pair = tile * 16 + r;
    const int col  = wave * 16 + r;

    const int ri = rowi[pair];
    const int ci = coli[pair];
    const float* zr = z + (size_t)ri * 64 + half * 2;
    const float* zc = z + (size_t)ci * 64 + half * 2;

    v8f acc = {};
#pragma unroll
    for (int k0 = 0; k0 < 64; k0 += 4) {   // concat dims [0,64): z[row]
        const int kk = k0 + half * 2;
        v2f a, b;
        a.x = zr[k0];
        a.y = zr[k0 + 1];
        b.x = Wd1[kk * 64 + col];
        b.y = Wd1[(kk + 1) * 64 + col];
        acc = __builtin_amdgcn_wmma_f32_16x16x4_f32(false, a, false, b,
                                                    (short)0, acc, false, false);
    }
#pragma unroll
    for (int k0 = 0; k0 < 64; k0 += 4) {   // concat dims [64,128): z[col]
        const int kk = 64 + k0 + half * 2;
        v2f a, b;
        a.x = zc[k0];
        a.y = zc[k0 + 1];
        b.x = Wd1[kk * 64 + col];
        b.y = Wd1[(kk + 1) * 64 + col];
        acc = __builtin_amdgcn_wmma_f32_16x16x4_f32(false, a, false, b,
                                                    (short)0, acc, false, false);
    }

    const float bias = bd1[col];
#pragma unroll
    for (int i = 0; i < 8; ++i) {
        const int m = i + half * 8;
        hbuf[m][col] = fmaxf(acc[i] + bias, 0.0f);
    }
    __syncthreads();

    if (threadIdx.x < 16) {
        float s = bd2[0];
#pragma unroll 8
        for (int j = 0; j < 64; ++j) s += hbuf[threadIdx.x][j] * Wd2[j];
        out[(size_t)tile * 16 + threadIdx.x] = s;
    }
}

// ---------------------------------------------------------------------------
extern "C" void kernel_launch(void* const* d_in, const int* in_sizes, int n_in,
                              void* d_out, int out_size, void* d_ws, size_t ws_size,
                              hipStream_t stream) {
    const float* x    = (const float*)d_in[0];
    const int*   eidx = (const int*)  d_in[1];
    const int*   lidx = (const int*)  d_in[2];
    const float* W1l  = (const float*)d_in[3];
    const float* b1l  = (const float*)d_in[4];
    const float* W1r  = (const float*)d_in[5];
    const float* W2l  = (const float*)d_in[6];
    const float* b2l  = (const float*)d_in[7];
    const float* W2r  = (const float*)d_in[8];
    const float* Wd1  = (const float*)d_in[9];
    const float* bd1  = (const float*)d_in[10];
    const float* Wd2  = (const float*)d_in[11];
    const float* bd2  = (const float*)d_in[12];

    const int* src = eidx;
    const int* dst = eidx + N_EDGES;
    const int* row = lidx;
    const int* col = lidx + N_LABEL;

    // workspace layout: agg[N*64] | cnt[N] | z1[N*64] | z2[N*64]
    float* agg = (float*)d_ws;
    float* cnt = agg + (size_t)N_NODES * 64;
    float* z1  = cnt + N_NODES;
    float* z2  = z1 + (size_t)N_NODES * 64;

    const int edgeBlocks = (N_EDGES * 32 + 255) / 256;   // 1 wave / edge
    const int nodeTiles  = N_NODES / 16;                 // 6250
    const int labelTiles = N_LABEL / 16;                 // 31250

    // ---- layer 1 ----
    zero_f32<<<8192, 256, 0, stream>>>(agg, (long)N_NODES * 64 + N_NODES);
    edge_scatter<<<edgeBlocks, 256, 0, stream>>>(x, src, dst, agg, cnt,
                                                 N_EDGES, 1);
    sage_combine<<<nodeTiles, 128, 0, stream>>>(agg, cnt, x, W1l, b1l, W1r,
                                                z1, 1);

    // ---- layer 2 (cnt is unchanged, reuse it) ----
    zero_f32<<<8192, 256, 0, stream>>>(agg, (long)N_NODES * 64);
    edge_scatter<<<edgeBlocks, 256, 0, stream>>>(z1, src, dst, agg, cnt,
                                                 N_EDGES, 0);
    sage_combine<<<nodeTiles, 128, 0, stream>>>(agg, cnt, z1, W2l, b2l, W2r,
                                                z2, 0);

    // ---- decoder ----
    link_decoder<<<labelTiles, 128, 0, stream>>>(z2, row, col, Wd1, bd1,
                                                 Wd2, bd2, (float*)d_out);
}